// LocalPoolPointnet_64759516889901
// MI455X (gfx1250) — compile-verified
//
#include <hip/hip_runtime.h>

// ---------------------------------------------------------------------------
// LocalPoolPointnet for MI455X (gfx1250, wave32).
// GEMMs use V_WMMA_F32_16X16X4_F32 (full-precision f32 WMMA).
// segment_max uses order-preserving float->u32 keys + global_atomic_max_u32.
// ---------------------------------------------------------------------------

typedef __attribute__((ext_vector_type(2))) float v2f;
typedef __attribute__((ext_vector_type(8))) float v8f;

#define BATCH 8
#define TPTS  100000
#define BT    (BATCH * TPTS)     // 800000 points
#define HID   32
#define X2H   64
#define R3    32768              // 32^3
#define SEGN  (BATCH * R3)       // 262144 segments

// workspace byte offsets (256B aligned)
#define OFF_NET   0u            // BT*32 f32          = 102,400,000
#define OFF_IDX   102400000u    // BT int             =   3,200,000
#define OFF_SEG   105600000u    // SEGN*32 u32        =  33,554,432
#define OFF_SUM   139154432u    // SEGN*32 f32        =  33,554,432
#define OFF_CNT   172708864u    // SEGN f32           =   1,048,576

// ---- order-preserving float <-> u32 key (for atomic max on floats) --------
__device__ __forceinline__ unsigned fkey(float f) {
    unsigned u = __float_as_uint(f);
    return (u & 0x80000000u) ? ~u : (u | 0x80000000u);
}
__device__ __forceinline__ float fdec(unsigned k) {
    unsigned u = (k & 0x80000000u) ? (k & 0x7fffffffu) : ~k;
    return __uint_as_float(u);
}

// ---- WMMA K-accumulation: acc += A(16xK) * Bt(K x 16 cols n0..n0+15) ------
// A:   LDS, row-major [16][lda], lane reads row=(lane&15), K pair at 2*half.
// Bt:  LDS, weights transposed [n][K] so each lane reads a contiguous pair.
// Fragment layouts per CDNA5 ISA 7.12.2 (f32 16x16x4).
template <int K, int LDA, bool RELU>
__device__ __forceinline__ v8f mm_acc(const float* __restrict__ A,
                                      const float* __restrict__ Bt,
                                      v8f acc, int row, int half, int n0) {
    const float* ar = A + row * LDA + 2 * half;
    const float* br = Bt + (n0 + row) * K + 2 * half;
#pragma unroll
    for (int k0 = 0; k0 < K; k0 += 4) {
        float a0 = ar[k0], a1 = ar[k0 + 1];
        if (RELU) { a0 = fmaxf(a0, 0.f); a1 = fmaxf(a1, 0.f); }
        v2f a; a.x = a0; a.y = a1;
        v2f b; b.x = br[k0]; b.y = br[k0 + 1];
        acc = __builtin_amdgcn_wmma_f32_16x16x4_f32(
            false, a, false, b, (short)0, acc, false, false);
    }
    return acc;
}

// ---- weight staging into LDS (transposed) ---------------------------------
__device__ __forceinline__ void stage_weights(
    int tid, const float* __restrict__ W0, const float* __restrict__ B0,
    const float* __restrict__ W1, const float* __restrict__ B1,
    const float* __restrict__ WS, float* w0t, float* w1t, float* wst,
    float* b0s, float* b1s) {
    for (int i = tid; i < 2048; i += 256) {
        int n = i >> 6, k = i & 63;               // w0t[n][k]
        w0t[i] = W0[k * 32 + n];
        wst[i] = WS[k * 32 + n];
    }
    for (int i = tid; i < 1024; i += 256) {
        int n = i >> 5, k = i & 31;
        w1t[i] = W1[k * 32 + n];
    }
    if (tid < 32) { b0s[tid] = B0[tid]; b1s[tid] = B1[tid]; }
}

// ---- one ResnetBlockFC for 16 points per wave -----------------------------
// xw: per-wave LDS [16][64] raw x.  hw: per-wave LDS [16][32] scratch.
// out0/out1: new net, C-fragment layout (tiles n=0..15 / n=16..31).
__device__ __forceinline__ void resblock_wmma(
    const float* __restrict__ xw, float* __restrict__ hw,
    const float* __restrict__ w0t, const float* __restrict__ w1t,
    const float* __restrict__ wst, const float* __restrict__ b0s,
    const float* __restrict__ b1s, int row, int half, v8f& out0, v8f& out1) {
    // h = relu( relu(x) @ W0 + b0 )
    v8f h0, h1;
    {
        float bb0 = b0s[row], bb1 = b0s[16 + row];
#pragma unroll
        for (int r = 0; r < 8; r++) { h0[r] = bb0; h1[r] = bb1; }
        h0 = mm_acc<64, 64, true>(xw, w0t, h0, row, half, 0);
        h1 = mm_acc<64, 64, true>(xw, w0t, h1, row, half, 16);
    }
#pragma unroll
    for (int r = 0; r < 8; r++) {
        hw[(r + 8 * half) * 32 + row]      = fmaxf(h0[r], 0.f);
        hw[(r + 8 * half) * 32 + 16 + row] = fmaxf(h1[r], 0.f);
    }
    __syncthreads();
    // net = x @ Ws + relu(h) @ W1 + b1
    v8f o0, o1;
    float c0 = b1s[row], c1 = b1s[16 + row];
#pragma unroll
    for (int r = 0; r < 8; r++) { o0[r] = c0; o1[r] = c1; }
    o0 = mm_acc<64, 64, false>(xw, wst, o0, row, half, 0);
    o0 = mm_acc<32, 32, false>(hw, w1t, o0, row, half, 0);
    o1 = mm_acc<64, 64, false>(xw, wst, o1, row, half, 16);
    o1 = mm_acc<32, 32, false>(hw, w1t, o1, row, half, 16);
    out0 = o0; out1 = o1;
}

// ===========================================================================
// Pass 0: voxel index + count, fc_pos, resblock 0  ->  net[BT][32]
// ===========================================================================
__global__ __launch_bounds__(256) void k_pass0(
    const float* __restrict__ p, const float* __restrict__ fpw,
    const float* __restrict__ fpb, const float* __restrict__ W0,
    const float* __restrict__ B0, const float* __restrict__ W1,
    const float* __restrict__ B1, const float* __restrict__ WS,
    float* __restrict__ net, int* __restrict__ idxb, float* __restrict__ cnt) {
    __shared__ __align__(16) float xs[128 * 64];
    __shared__ __align__(16) float hs[128 * 32];
    __shared__ float w0t[32 * 64], wst[32 * 64], w1t[32 * 32];
    __shared__ float b0s[32], b1s[32];
    __shared__ float fpws[3 * 64], fpbs[64];

    int tid = threadIdx.x;
    stage_weights(tid, W0, B0, W1, B1, WS, w0t, w1t, wst, b0s, b1s);
    for (int i = tid; i < 192; i += 256) fpws[i] = fpw[i];
    if (tid < 64) fpbs[tid] = fpb[tid];
    __syncthreads();

    int ptl = tid >> 1, hsel = tid & 1;
    int pt = blockIdx.x * 128 + ptl;
    float px = p[pt * 3 + 0], py = p[pt * 3 + 1], pz = p[pt * 3 + 2];
#pragma unroll
    for (int j = 0; j < 32; j++) {
        int jj = hsel * 32 + j;
        xs[ptl * 64 + jj] =
            fmaf(px, fpws[jj], fmaf(py, fpws[64 + jj],
                 fmaf(pz, fpws[128 + jj], fpbs[jj])));
    }
    if (!hsel) {
        float c0 = fminf(fmaxf(px, 0.f), 1.f - 1e-5f);
        float c1 = fminf(fmaxf(py, 0.f), 1.f - 1e-5f);
        float c2 = fminf(fmaxf(pz, 0.f), 1.f - 1e-5f);
        int x0 = (int)(c0 * 32.f), x1 = (int)(c1 * 32.f), x2 = (int)(c2 * 32.f);
        int bb = pt / TPTS;
        int fl = x0 + 32 * (x1 + 32 * x2) + bb * R3;
        idxb[pt] = fl;
        atomicAdd(&cnt[fl], 1.0f);
    }
    __syncthreads();

    int lane = tid & 31, wv = tid >> 5, row = lane & 15, half = lane >> 4;
    v8f o0, o1;
    resblock_wmma(xs + wv * 16 * 64, hs + wv * 16 * 32, w0t, w1t, wst, b0s,
                  b1s, row, half, o0, o1);
    int pbase = blockIdx.x * 128 + wv * 16;
#pragma unroll
    for (int r = 0; r < 8; r++) {
        int q = (pbase + r + 8 * half) * 32;
        net[q + row]      = o0[r];
        net[q + 16 + row] = o1[r];
    }
}

// ===========================================================================
// scatter_max: per point, 32x global_atomic_max_u32 into key buffer
// ===========================================================================
__global__ __launch_bounds__(256) void k_scatter(
    const float* __restrict__ net, const int* __restrict__ idxb,
    unsigned* __restrict__ seg) {
    int pt = blockIdx.x * 256 + threadIdx.x;
    int fl = idxb[pt];
    const float4* nr = (const float4*)(net + (size_t)pt * 32);
    unsigned* s = seg + (size_t)fl * 32;
#pragma unroll
    for (int q = 0; q < 8; q++) {
        float4 v = nr[q];
        atomicMax(&s[q * 4 + 0], fkey(v.x));
        atomicMax(&s[q * 4 + 1], fkey(v.y));
        atomicMax(&s[q * 4 + 2], fkey(v.z));
        atomicMax(&s[q * 4 + 3], fkey(v.w));
    }
}

// ---- shared body for the pooled resblock rounds ---------------------------
__device__ __forceinline__ void pool_build_x(
    int tid, int blk, const float* __restrict__ net,
    const unsigned* __restrict__ seg, const int* __restrict__ idxb,
    float* __restrict__ xs) {
    int ptl = tid >> 1, hsel = tid & 1;
    int pt = blk * 128 + ptl;
    if (!hsel) {                       // x[0:32] = net
        const float4* src = (const float4*)(net + (size_t)pt * 32);
        float4* dst = (float4*)(xs + ptl * 64);
#pragma unroll
        for (int q = 0; q < 8; q++) dst[q] = src[q];
    } else {                           // x[32:64] = pooled (decoded max keys)
        int fl = idxb[pt];
        const uint4* sp = (const uint4*)(seg + (size_t)fl * 32);
#pragma unroll
        for (int q = 0; q < 8; q++) {
            uint4 v = sp[q];
            float* d = xs + ptl * 64 + 32 + q * 4;
            d[0] = fdec(v.x); d[1] = fdec(v.y);
            d[2] = fdec(v.z); d[3] = fdec(v.w);
        }
    }
}

// ===========================================================================
// rounds 1..3: gather + concat + resblock -> net (in place)
// ===========================================================================
__global__ __launch_bounds__(256) void k_pool(
    const unsigned* __restrict__ seg, const int* __restrict__ idxb,
    const float* __restrict__ W0, const float* __restrict__ B0,
    const float* __restrict__ W1, const float* __restrict__ B1,
    const float* __restrict__ WS, float* __restrict__ net) {
    __shared__ __align__(16) float xs[128 * 64];
    __shared__ __align__(16) float hs[128 * 32];
    __shared__ float w0t[32 * 64], wst[32 * 64], w1t[32 * 32];
    __shared__ float b0s[32], b1s[32];

    int tid = threadIdx.x;
    stage_weights(tid, W0, B0, W1, B1, WS, w0t, w1t, wst, b0s, b1s);
    pool_build_x(tid, blockIdx.x, net, seg, idxb, xs);
    __syncthreads();

    int lane = tid & 31, wv = tid >> 5, row = lane & 15, half = lane >> 4;
    v8f o0, o1;
    resblock_wmma(xs + wv * 16 * 64, hs + wv * 16 * 32, w0t, w1t, wst, b0s,
                  b1s, row, half, o0, o1);
    int pbase = blockIdx.x * 128 + wv * 16;
#pragma unroll
    for (int r = 0; r < 8; r++) {
        int q = (pbase + r + 8 * half) * 32;
        net[q + row]      = o0[r];
        net[q + 16 + row] = o1[r];
    }
}

// ===========================================================================
// round 4: resblock + fc_c + scatter-add into sums
// ===========================================================================
__global__ __launch_bounds__(256) void k_pool_final(
    const unsigned* __restrict__ seg, const int* __restrict__ idxb,
    const float* __restrict__ W0, const float* __restrict__ B0,
    const float* __restrict__ W1, const float* __restrict__ B1,
    const float* __restrict__ WS, const float* __restrict__ FCW,
    const float* __restrict__ FCB, const float* __restrict__ net,
    float* __restrict__ sums) {
    __shared__ __align__(16) float xs[128 * 64];
    __shared__ __align__(16) float hs[128 * 32];
    __shared__ float w0t[32 * 64], wst[32 * 64], w1t[32 * 32];
    __shared__ float b0s[32], b1s[32];
    __shared__ float fct[32 * 32], fcbs[32];

    int tid = threadIdx.x;
    stage_weights(tid, W0, B0, W1, B1, WS, w0t, w1t, wst, b0s, b1s);
    for (int i = tid; i < 1024; i += 256) {
        int n = i >> 5, k = i & 31;
        fct[i] = FCW[k * 32 + n];
    }
    if (tid < 32) fcbs[tid] = FCB[tid];
    pool_build_x(tid, blockIdx.x, net, seg, idxb, xs);
    __syncthreads();

    int lane = tid & 31, wv = tid >> 5, row = lane & 15, half = lane >> 4;
    float* hw = hs + wv * 16 * 32;
    v8f o0, o1;
    resblock_wmma(xs + wv * 16 * 64, hw, w0t, w1t, wst, b0s, b1s, row, half,
                  o0, o1);
    __syncthreads();                    // hw (h values) dead, reuse for net
#pragma unroll
    for (int r = 0; r < 8; r++) {
        hw[(r + 8 * half) * 32 + row]      = o0[r];
        hw[(r + 8 * half) * 32 + 16 + row] = o1[r];
    }
    __syncthreads();
    // c = net @ fc_c_w + fc_c_b
    v8f s0, s1;
    float b0f = fcbs[row], b1f = fcbs[16 + row];
#pragma unroll
    for (int r = 0; r < 8; r++) { s0[r] = b0f; s1[r] = b1f; }
    s0 = mm_acc<32, 32, false>(hw, fct, s0, row, half, 0);
    s1 = mm_acc<32, 32, false>(hw, fct, s1, row, half, 16);

    int pbase = blockIdx.x * 128 + wv * 16;
#pragma unroll
    for (int r = 0; r < 8; r++) {
        int pt = pbase + r + 8 * half;
        int fl = idxb[pt];
        atomicAdd(&sums[(size_t)fl * 32 + row],      s0[r]);
        atomicAdd(&sums[(size_t)fl * 32 + 16 + row], s1[r]);
    }
}

// ===========================================================================
// finalize: grid[b][ch][vox] = sums / max(cnt,1)
// ===========================================================================
__global__ __launch_bounds__(256) void k_fin(
    const float* __restrict__ sums, const float* __restrict__ cnt,
    float* __restrict__ out) {
    int g = blockIdx.x * 256 + threadIdx.x;   // g = s*32 + ch
    int s = g >> 5, ch = g & 31;
    float c = cnt[s];
    float v = sums[g] / fmaxf(c, 1.0f);
    int b = s >> 15, vox = s & (R3 - 1);
    out[(((size_t)((b << 5) + ch)) << 15) + vox] = v;
}

// ===========================================================================
extern "C" void kernel_launch(void* const* d_in, const int* in_sizes, int n_in,
                              void* d_out, int out_size, void* d_ws,
                              size_t ws_size, hipStream_t stream) {
    const float* p   = (const float*)d_in[0];
    const float* fpw = (const float*)d_in[1];
    const float* fpb = (const float*)d_in[2];
    const float* w0  = (const float*)d_in[3];
    const float* b0  = (const float*)d_in[4];
    const float* w1  = (const float*)d_in[5];
    const float* b1  = (const float*)d_in[6];
    const float* ws  = (const float*)d_in[7];
    const float* fcw = (const float*)d_in[8];
    const float* fcb = (const float*)d_in[9];
    float* out = (float*)d_out;

    char* W = (char*)d_ws;
    float*    net  = (float*)(W + OFF_NET);
    int*      idxb = (int*)(W + OFF_IDX);
    unsigned* seg  = (unsigned*)(W + OFF_SEG);
    float*    sums = (float*)(W + OFF_SUM);
    float*    cnt  = (float*)(W + OFF_CNT);

    hipMemsetAsync(sums, 0, (size_t)SEGN * 32 * 4, stream);
    hipMemsetAsync(cnt, 0, (size_t)SEGN * 4, stream);

    k_pass0<<<BT / 128, 256, 0, stream>>>(p, fpw, fpb, w0, b0, w1, b1, ws,
                                          net, idxb, cnt);
    for (int i = 1; i < 5; i++) {
        hipMemsetAsync(seg, 0, (size_t)SEGN * 32 * 4, stream);
        k_scatter<<<BT / 256, 256, 0, stream>>>(net, idxb, seg);
        const float* W0 = w0 + i * 2048;
        const float* B0 = b0 + i * 32;
        const float* W1 = w1 + i * 1024;
        const float* B1 = b1 + i * 32;
        const float* WS = ws + i * 2048;
        if (i < 4)
            k_pool<<<BT / 128, 256, 0, stream>>>(seg, idxb, W0, B0, W1, B1,
                                                 WS, net);
        else
            k_pool_final<<<BT / 128, 256, 0, stream>>>(seg, idxb, W0, B0, W1,
                                                       B1, WS, fcw, fcb, net,
                                                       sums);
    }
    k_fin<<<(SEGN * 32) / 256, 256, 0, stream>>>(sums, cnt, out);
}